// HierPoolStage_64888365907994
// MI455X (gfx1250) — compile-verified
//
#include <hip/hip_runtime.h>
#include <hip/hip_bf16.h>

#define FLT_BIG 3.402823466e+38f
#define RADIUS2 0.01f
#define K_LOCAL 32
#define CAP 2048

typedef __attribute__((ext_vector_type(2))) float v2f;
typedef __attribute__((ext_vector_type(8))) float v8f;

// ---------------------------------------------------------------------------
// Kernel 1: Farthest point sampling. Single 1024-thread workgroup (32 waves),
// dist[] in global scratch (L2 resident), LDS tree argmax per step.
// Matches jnp.argmax tie-break (lowest index wins).
// ---------------------------------------------------------------------------
__global__ __launch_bounds__(1024)
void fps_kernel(const float* __restrict__ pos, float* __restrict__ dist,
                int* __restrict__ cent_idx, float* __restrict__ cents,
                int N, int M) {
  const int tid = threadIdx.x;
  __shared__ int   s_cur;
  __shared__ float s_cp[3];
  __shared__ float s_rv[1024];
  __shared__ int   s_ri[1024];

  for (int i = tid; i < N; i += 1024) dist[i] = FLT_BIG;
  if (tid == 0) s_cur = 0;
  __syncthreads();

  for (int it = 0; it < M; ++it) {
    if (tid == 0) {
      int cur = s_cur;
      cent_idx[it] = cur;
      float x = pos[cur * 3 + 0];
      float y = pos[cur * 3 + 1];
      float z = pos[cur * 3 + 2];
      s_cp[0] = x; s_cp[1] = y; s_cp[2] = z;
      cents[it * 3 + 0] = x;
      cents[it * 3 + 1] = y;
      cents[it * 3 + 2] = z;
    }
    __syncthreads();
    const float cx = s_cp[0], cy = s_cp[1], cz = s_cp[2];
    float bv = -FLT_BIG;
    int   bi = 0x7fffffff;
    for (int i = tid; i < N; i += 1024) {
      float dx = pos[i * 3 + 0] - cx;
      float dy = pos[i * 3 + 1] - cy;
      float dz = pos[i * 3 + 2] - cz;
      float d2 = dx * dx + dy * dy + dz * dz;
      float dd = fminf(dist[i], d2);
      dist[i] = dd;
      if (dd > bv) { bv = dd; bi = i; }   // strict > keeps lowest index
    }
    s_rv[tid] = bv;
    s_ri[tid] = bi;
    __syncthreads();
    for (int off = 512; off >= 1; off >>= 1) {
      if (tid < off) {
        float ov = s_rv[tid + off];
        int   oi = s_ri[tid + off];
        if (ov > s_rv[tid] || (ov == s_rv[tid] && oi < s_ri[tid])) {
          s_rv[tid] = ov; s_ri[tid] = oi;
        }
      }
      __syncthreads();
    }
    if (tid == 0) s_cur = s_ri[0];
    __syncthreads();
  }
}

// ---------------------------------------------------------------------------
// Kernel 2: Ball query + k-smallest selection + fused feature max-pool.
// One 256-thread block per centroid. Candidates collected in LDS; 32
// sequential argmin reductions with (d2, index) lexicographic tie-break.
// Pad-with-nearest is a no-op under max-pool, so padding uses sel[0].
// ---------------------------------------------------------------------------
__global__ __launch_bounds__(256)
void bq_pool_kernel(const float* __restrict__ pos,
                    const float* __restrict__ fs,   // (N,C,1)
                    const float* __restrict__ fv,   // (N,C,3)
                    const int* __restrict__ cent_idx,
                    float* __restrict__ pooled_s,   // (M,C)
                    float* __restrict__ pooled_v,   // 3 planes of (M,C)
                    int N, int C, int M) {
  const int m   = blockIdx.x;
  const int tid = threadIdx.x;
  __shared__ float s_cd2[CAP];
  __shared__ int   s_cidx[CAP];
  __shared__ int   s_cnt;
  __shared__ int   s_sel[K_LOCAL];
  __shared__ float s_rv[256];
  __shared__ int   s_ri[256];
  __shared__ int   s_rp[256];
  __shared__ float s_c[3];

  if (tid == 0) {
    s_cnt = 0;
    int ci = cent_idx[m];
    s_c[0] = pos[ci * 3 + 0];
    s_c[1] = pos[ci * 3 + 1];
    s_c[2] = pos[ci * 3 + 2];
  }
  __syncthreads();
  const float cx = s_c[0], cy = s_c[1], cz = s_c[2];

  for (int i = tid; i < N; i += 256) {
    float dx = pos[i * 3 + 0] - cx;
    float dy = pos[i * 3 + 1] - cy;
    float dz = pos[i * 3 + 2] - cz;
    float d2 = dx * dx + dy * dy + dz * dz;
    if (d2 <= RADIUS2) {
      int p = atomicAdd(&s_cnt, 1);
      if (p < CAP) { s_cd2[p] = d2; s_cidx[p] = i; }
    }
  }
  __syncthreads();
  const int ncand = min(s_cnt, CAP);

  for (int s = 0; s < K_LOCAL; ++s) {
    float bv = FLT_BIG;
    int   bi = 0x7fffffff;
    int   bp = -1;
    for (int p = tid; p < ncand; p += 256) {
      float v  = s_cd2[p];
      int   ix = s_cidx[p];
      if (v < bv || (v == bv && ix < bi)) { bv = v; bi = ix; bp = p; }
    }
    s_rv[tid] = bv; s_ri[tid] = bi; s_rp[tid] = bp;
    __syncthreads();
    for (int off = 128; off >= 1; off >>= 1) {
      if (tid < off) {
        float ov = s_rv[tid + off];
        int   oi = s_ri[tid + off];
        if (ov < s_rv[tid] || (ov == s_rv[tid] && oi < s_ri[tid])) {
          s_rv[tid] = ov; s_ri[tid] = oi; s_rp[tid] = s_rp[tid + off];
        }
      }
      __syncthreads();
    }
    if (tid == 0) {
      if (s < ncand) {
        int p = s_rp[0];
        s_sel[s] = s_cidx[p];
        s_cd2[p] = FLT_BIG;     // consume
      } else {
        s_sel[s] = s_sel[0];    // pad with nearest (no-op under max)
      }
    }
    __syncthreads();
  }

  // fused gather + max-pool: 4 d-slots per channel (0=scalar, 1..3=vector)
  for (int j = tid; j < C * 4; j += 256) {
    const int c = j >> 2;
    const int d = j & 3;
    float mx = -FLT_BIG;
    #pragma unroll
    for (int t = 0; t < K_LOCAL; ++t) {
      int n = s_sel[t];
      float v = (d == 0) ? fs[(size_t)n * C + c]
                         : fv[((size_t)n * C + c) * 3 + (d - 1)];
      mx = fmaxf(mx, v);
    }
    if (d == 0) pooled_s[(size_t)m * C + c] = mx;
    else        pooled_v[(size_t)(d - 1) * M * C + (size_t)m * C + c] = mx;
  }
}

// ---------------------------------------------------------------------------
// Kernel 3: Channel mixer via native fp32 WMMA (V_WMMA_F32_16X16X4_F32).
// out[m,o,d] = sum_c pooled[m,c,d] * W[c,o] + b[o], full fp32 precision.
// One wave per 16x16 output tile; 8 waves per block.
// A layout: lane = m%16 + 16*(k/2), vgpr = k%2. B mirrors with n. D: element
// (m,n) -> vgpr m%8, lane n + 16*(m/8).
// ---------------------------------------------------------------------------
__global__ __launch_bounds__(256)
void mix_wmma_kernel(const float* __restrict__ pooled_s,
                     const float* __restrict__ pooled_v,
                     const float* __restrict__ Ws, const float* __restrict__ Wv,
                     const float* __restrict__ bs, const float* __restrict__ bv,
                     float* __restrict__ outS,   // (M,C)
                     float* __restrict__ outV,   // (M,C,3)
                     int M, int C) {
  const int lane = threadIdx.x & 31;
  const int wave = threadIdx.x >> 5;
  const int tiles_o = C >> 4;
  const int tiles_m = M >> 4;
  const int per_d   = tiles_m * tiles_o;
  const int tile    = blockIdx.x * (blockDim.x >> 5) + wave;
  if (tile >= per_d * 4) return;   // wave-uniform; EXEC stays all-1s for WMMA

  const int d   = tile / per_d;
  const int rem = tile % per_d;
  const int m0  = (rem / tiles_o) << 4;
  const int o0  = (rem % tiles_o) << 4;

  const float* A    = (d == 0) ? (pooled_s + (size_t)m0 * C)
                               : (pooled_v + (size_t)(d - 1) * M * C + (size_t)m0 * C);
  const float* B    = (d == 0) ? Ws : Wv;
  const float* bias = (d == 0) ? bs : bv;

  const int ml = lane & 15;   // m within tile for A; n within tile for B/D
  const int kh = lane >> 4;   // which k-pair half

  v8f acc = {};
  for (int k0 = 0; k0 < C; k0 += 4) {
    const int ka = k0 + 2 * kh;
    v2f a = *(const v2f*)(A + (size_t)ml * C + ka);
    v2f b;
    b.x = B[(size_t)ka * C + o0 + ml];
    b.y = B[(size_t)(ka + 1) * C + o0 + ml];
    // 8 args: (neg_a, A, neg_b, B, c_mod, C, reuse_a, reuse_b)
    acc = __builtin_amdgcn_wmma_f32_16x16x4_f32(
        false, a, false, b, (short)0, acc, false, false);
  }

  const int   n  = o0 + ml;
  const float bb = bias[n];
  #pragma unroll
  for (int v = 0; v < 8; ++v) {
    const int mm  = m0 + v + 8 * kh;
    const float val = acc[v] + bb;
    if (d == 0) outS[(size_t)mm * C + n] = val;
    else        outV[((size_t)mm * C + n) * 3 + (d - 1)] = val;
  }
}

// ---------------------------------------------------------------------------
extern "C" void kernel_launch(void* const* d_in, const int* in_sizes, int n_in,
                              void* d_out, int out_size, void* d_ws, size_t ws_size,
                              hipStream_t stream) {
  const float* pos = (const float*)d_in[0];
  const float* fs  = (const float*)d_in[1];
  const float* fv  = (const float*)d_in[2];
  const float* Ws  = (const float*)d_in[3];
  const float* Wv  = (const float*)d_in[4];
  const float* bs  = (const float*)d_in[5];
  const float* bv  = (const float*)d_in[6];

  const int N = in_sizes[0] / 3;
  const int C = in_sizes[1] / N;                 // feat_scalar is (N, C, 1)
  const int M = (N < 2048) ? N : 2048;

  float* out   = (float*)d_out;
  float* cents = out;                            // (M,3)
  float* outS  = out + (size_t)M * 3;            // (M,C,1)
  float* outV  = outS + (size_t)M * C;           // (M,C,3)

  // scratch layout
  int*   cent_idx = (int*)d_ws;                  // M ints
  float* wsf      = (float*)d_ws;
  float* dist     = wsf + 4096;                  // N floats
  float* pooled_s = wsf + 4096 + 65536;          // M*C floats
  float* pooled_v = pooled_s + (size_t)M * C;    // 3*M*C floats (plane-major)

  fps_kernel<<<1, 1024, 0, stream>>>(pos, dist, cent_idx, cents, N, M);
  bq_pool_kernel<<<M, 256, 0, stream>>>(pos, fs, fv, cent_idx,
                                        pooled_s, pooled_v, N, C, M);

  const int per_d = (M / 16) * (C / 16);
  const int total = per_d * 4;
  mix_wmma_kernel<<<(total + 7) / 8, 256, 0, stream>>>(
      pooled_s, pooled_v, Ws, Wv, bs, bv, outS, outV, M, C);
}